// Model_66254165508715
// MI455X (gfx1250) — compile-verified
//
#include <hip/hip_runtime.h>
#include <hip/hip_bf16.h>

// ---------------------------------------------------------------------------
// Model: 4-layer relative-position GNN + MLP head, fp32, B=8, N=512.
// Key algebraic collapse: msum[b,i] = mask_i * ( cnt_b*(ai_i + bm)
//                                   + sum_j mask_j*aj_j
//                                   + sum_j mask_j*We[N-1+i-j] )
// All dense GEMMs (4096 x smallK x smallN) run on V_WMMA_F32_16X16X4_F32,
// 4 row-tiles per wave (B-fragment reuse), strides/K as template constants
// so addressing folds into immediate offsets and load clauses.
// ---------------------------------------------------------------------------

typedef float v2f __attribute__((ext_vector_type(2)));
typedef float v8f __attribute__((ext_vector_type(8)));

#define RB    8
#define NN    512
#define RTOT  (RB * NN)      // 4096 rows
#define EE    1023           // 2N-1
#define MM    16
#define II    64
#define JJ    128

// ---------------------------------------------------------------------------
// fp32 WMMA GEMM, ROWBLK row-tiles per wave:
//   out[r0:r0+16*ROWBLK, n0:n0+16] = X @ W (+bias)(relu),  K % 4 == 0
// A-frag (16x4):  lanes 0-15 row=lane {K0,K1}; lanes 16-31 row=lane-16 {K2,K3}
// B-frag (4x16):  lanes 0-15 col=lane {K0,K1}; lanes 16-31 col=lane-16 {K2,K3}
// C/D  (16x16):   vgpr v: lanes 0-15 row v, lanes 16-31 row v+8
// ---------------------------------------------------------------------------
template <int ROWBLK, int LDX, int LDW, int LDO, int K, int NTILES, int RELU>
__global__ void gemm16_wmma(const float* __restrict__ X,
                            const float* __restrict__ W,
                            const float* __restrict__ bias,
                            float* __restrict__ out)
{
    constexpr int rGroups = (RTOT / 16) / ROWBLK;
    int wave = blockIdx.x * (blockDim.x >> 5) + (threadIdx.x >> 5);
    if (wave >= rGroups * NTILES) return;      // whole-wave exit: EXEC stays all-ones
    int rg = wave / NTILES, nt = wave % NTILES;
    int r0 = rg * (16 * ROWBLK), n0 = nt << 4;

    int lane = threadIdx.x & 31;
    int half = lane >> 4;                      // 0: K pair {k0,k0+1}, 1: {k0+2,k0+3}
    int l15  = lane & 15;

    const float* xbase = X + (size_t)(r0 + l15) * LDX;   // tile t at +16*t*LDX (const)
    const float* wbase = W + n0 + l15;                   // k-step at +ka*LDW (const)

    v8f acc[ROWBLK];
#pragma unroll
    for (int t = 0; t < ROWBLK; ++t) acc[t] = (v8f){};

#pragma unroll 4
    for (int k0 = 0; k0 < K; k0 += 4) {
        int ka = k0 + 2 * half;
        // pull next B line toward the WGP while these WMMAs run
        __builtin_prefetch(wbase + (size_t)(ka + 4) * LDW, 0, 1);
        v2f b;
        b.x = wbase[(size_t)ka * LDW];
        b.y = wbase[(size_t)(ka + 1) * LDW];
#pragma unroll
        for (int t = 0; t < ROWBLK; ++t) {
            v2f a;
            a.x = xbase[(size_t)(16 * t) * LDX + ka];
            a.y = xbase[(size_t)(16 * t) * LDX + ka + 1];
            acc[t] = __builtin_amdgcn_wmma_f32_16x16x4_f32(
                         false, a, false, b, (short)0, acc[t], false, false);
        }
    }

    float bv = bias ? bias[n0 + l15] : 0.0f;
#pragma unroll
    for (int t = 0; t < ROWBLK; ++t) {
        int row = r0 + 16 * t + 8 * half;
#pragma unroll
        for (int v = 0; v < 8; ++v) {
            float val = acc[t][v] + bv;
            if (RELU) val = fmaxf(val, 0.0f);
            out[(size_t)(row + v) * LDO + n0 + l15] = val;
        }
    }
}

// ---------------------------------------------------------------------------
// Build h0 (R x 48, cols 0..29 valid): [frames7 | features | t/100]
// ---------------------------------------------------------------------------
__global__ void prep_h0(const float* __restrict__ frames7,
                        const float* __restrict__ features,
                        const int* __restrict__ tptr,
                        float* __restrict__ h0)
{
    int idx = blockIdx.x * blockDim.x + threadIdx.x;   // over R*32
    if (idx >= RTOT * 32) return;
    int r = idx >> 5, c = idx & 31;
    float v;
    if      (c < 7)   v = frames7[(size_t)r * 7 + c];
    else if (c < 29)  v = features[(size_t)r * 22 + (c - 7)];
    else if (c == 29) v = (float)(*tptr) * 0.01f;
    else return;                                       // cols 30..47: msum / zero-weight pad
    h0[(size_t)r * 48 + c] = v;
}

// ---------------------------------------------------------------------------
// Zero-padded weight pack: dst[rowsPad x cols] = src[row0+rp, c] or 0
// ---------------------------------------------------------------------------
__global__ void pack_pad(const float* __restrict__ src, float* __restrict__ dst,
                         int row0, int rows, int rowsPad, int cols)
{
    int idx = blockIdx.x * blockDim.x + threadIdx.x;
    if (idx >= rowsPad * cols) return;
    int rp = idx / cols, c = idx - rp * cols;
    dst[idx] = (rp < rows) ? src[(size_t)(row0 + rp) * cols + c] : 0.0f;
}

// ---------------------------------------------------------------------------
// Per-batch reductions: cnt_b = sum_i mask, Sb[b,m] = sum_i mask_i * aj[b,i,m]
// ---------------------------------------------------------------------------
__global__ void reduce_b_kernel(const float* __restrict__ aj,
                                const unsigned char* __restrict__ mask,
                                float* __restrict__ Sb, float* __restrict__ cnt)
{
    int b = blockIdx.x;
    int t = threadIdx.x;
    int m = t & 15, c = t >> 4;
    __shared__ float sred[16][17];
    __shared__ float scnt[16];
    float s = 0.0f, cn = 0.0f;
    for (int i = c * 32; i < c * 32 + 32; ++i) {
        float mk = mask[b * NN + i] ? 1.0f : 0.0f;
        s += mk * aj[((size_t)b * NN + i) * MM + m];
        if (m == 0) cn += mk;
    }
    sred[c][m] = s;
    if (m == 0) scnt[c] = cn;
    __syncthreads();
    if (c == 0) {
        float tot = 0.0f;
        for (int cc = 0; cc < 16; ++cc) tot += sred[cc][m];
        Sb[b * MM + m] = tot;
        if (m == 0) {
            float ct = 0.0f;
            for (int cc = 0; cc < 16; ++cc) ct += scnt[cc];
            cnt[b] = ct;
        }
    }
}

// ---------------------------------------------------------------------------
// Fused wesum correlation + msum finalize, writing msum into hout[:, colOff..+15].
// ---------------------------------------------------------------------------
__global__ void msum_kernel(const float* __restrict__ We,     // E x 16
                            const float* __restrict__ ai,     // R x 16
                            const float* __restrict__ bm,     // 16
                            const float* __restrict__ Sb,     // B x 16
                            const float* __restrict__ cnt,    // B
                            const unsigned char* __restrict__ mask,
                            float* __restrict__ hout, int ldh, int colOff)
{
    __shared__ float mk[NN];
    __shared__ float weS[527 * 16];
    int b  = blockIdx.x >> 5;
    int i0 = (blockIdx.x & 31) << 4;
    int t  = threadIdx.x;                       // 256 threads

    for (int j = t; j < NN; j += 256)
        mk[j] = mask[b * NN + j] ? 1.0f : 0.0f;
    for (int idx = t; idx < 527 * 16; idx += 256)
        weS[idx] = We[(size_t)i0 * 16 + idx];   // rows i0 .. i0+526 (max = E-1)
    __syncthreads();

    int m = t & 15, il = t >> 4;
    const float* wrow = &weS[(511 + il) * 16 + m];
    float acc = 0.0f;
    for (int j = 0; j < NN; ++j)
        acc += mk[j] * wrow[-j * 16];           // We[N-1 + i - j]

    int    i = i0 + il;
    size_t r = (size_t)b * NN + i;
    float  v = mk[i] * (cnt[b] * (ai[r * MM + m] + bm[m]) + Sb[b * MM + m] + acc);
    hout[r * ldh + colOff + m] = v;
}

// ---------------------------------------------------------------------------
// Head tail: o = z @ Wt2 + bt2 (128 -> 7), quaternion-normalize o[0:4].
// ---------------------------------------------------------------------------
__global__ void head_final(const float* __restrict__ z,
                           const float* __restrict__ Wt2,
                           const float* __restrict__ bt2,
                           float* __restrict__ out)
{
    int r = blockIdx.x * blockDim.x + threadIdx.x;
    if (r >= RTOT) return;
    float o[7];
#pragma unroll
    for (int c = 0; c < 7; ++c) o[c] = bt2[c];
    const float* zr = z + (size_t)r * JJ;
    for (int k = 0; k < JJ; ++k) {
        float zv = zr[k];
        const float* w = Wt2 + k * 7;
#pragma unroll
        for (int c = 0; c < 7; ++c) o[c] += zv * w[c];
    }
    float inv = 1.0f / sqrtf(o[0]*o[0] + o[1]*o[1] + o[2]*o[2] + o[3]*o[3]);
    float* op = out + (size_t)r * 7;
    op[0] = o[0] * inv; op[1] = o[1] * inv;
    op[2] = o[2] * inv; op[3] = o[3] * inv;
    op[4] = o[4]; op[5] = o[5]; op[6] = o[6];
}

// ---------------------------------------------------------------------------
// Orchestration
// ---------------------------------------------------------------------------
extern "C" void kernel_launch(void* const* d_in, const int* in_sizes, int n_in,
                              void* d_out, int out_size, void* d_ws, size_t ws_size,
                              hipStream_t stream)
{
    const float* frames7  = (const float*)d_in[0];
    const float* features = (const float*)d_in[1];
    const float* Wm[4] = { (const float*)d_in[2],  (const float*)d_in[6],
                           (const float*)d_in[10], (const float*)d_in[14] };
    const float* bm[4] = { (const float*)d_in[3],  (const float*)d_in[7],
                           (const float*)d_in[11], (const float*)d_in[15] };
    const float* Wf[4] = { (const float*)d_in[4],  (const float*)d_in[8],
                           (const float*)d_in[12], (const float*)d_in[16] };
    const float* bf[4] = { (const float*)d_in[5],  (const float*)d_in[9],
                           (const float*)d_in[13], (const float*)d_in[17] };
    const float* Wt1 = (const float*)d_in[18];
    const float* bt1 = (const float*)d_in[19];
    const float* Wt2 = (const float*)d_in[20];
    const float* bt2 = (const float*)d_in[21];
    const unsigned char* mask = (const unsigned char*)d_in[22];
    const int* tptr = (const int*)d_in[23];

    float* ws = (float*)d_ws;
    size_t off = 0;
    float* h0   = ws + off; off += (size_t)RTOT * 48;
    float* hA   = ws + off; off += (size_t)RTOT * 80;
    float* hB   = ws + off; off += (size_t)RTOT * 80;
    float* ai   = ws + off; off += (size_t)RTOT * MM;
    float* aj   = ws + off; off += (size_t)RTOT * MM;
    float* Sb   = ws + off; off += RB * MM;
    float* cnt  = ws + off; off += RB;
    float* zbuf = ws + off; off += (size_t)RTOT * JJ;
    float* W1iP = ws + off; off += 32 * 16;
    float* W1jP = ws + off; off += 32 * 16;
    float* W1fP = ws + off; off += 48 * 64;

    // h0 and padded layer-1 weights
    prep_h0<<<(RTOT * 32 + 255) / 256, 256, 0, stream>>>(frames7, features, tptr, h0);
    pack_pad<<<(32 * 16 + 255) / 256, 256, 0, stream>>>(Wm[0], W1iP,  0, 30, 32, 16);
    pack_pad<<<(32 * 16 + 255) / 256, 256, 0, stream>>>(Wm[0], W1jP, 30, 30, 32, 16);
    pack_pad<<<(48 * 64 + 255) / 256, 256, 0, stream>>>(Wf[0], W1fP,  0, 46, 48, 64);

    // ---------------- Layer 1 (Hd=30 pad 32; Kf=46 pad 48; X=h0 ld48) ------
    gemm16_wmma<4, 48, 16, 16, 32, 1, 0><<<8, 256, 0, stream>>>(h0, W1iP, nullptr, ai);
    gemm16_wmma<4, 48, 16, 16, 32, 1, 0><<<8, 256, 0, stream>>>(h0, W1jP, nullptr, aj);
    reduce_b_kernel<<<RB, 256, 0, stream>>>(aj, mask, Sb, cnt);
    msum_kernel<<<RB * 32, 256, 0, stream>>>(Wm[0] + 60 * 16, ai, bm[0], Sb, cnt, mask, h0, 48, 30);
    gemm16_wmma<4, 48, 64, 80, 48, 4, 1><<<32, 256, 0, stream>>>(h0, W1fP, bf[0], hA);

    // ---------------- Layers 2-4 (Hd=64; Kf=80; ld80) ----------------------
    const float* Xl[3]    = { hA, hB, hA };
    float*       Hout[3]  = { hB, hA, hB };
    for (int l = 0; l < 3; ++l) {
        const float* X  = Xl[l];
        const float* Wl = Wm[l + 1];
        gemm16_wmma<4, 80, 16, 16, 64, 1, 0><<<8, 256, 0, stream>>>(X, Wl,           nullptr, ai);
        gemm16_wmma<4, 80, 16, 16, 64, 1, 0><<<8, 256, 0, stream>>>(X, Wl + 64 * 16, nullptr, aj);
        reduce_b_kernel<<<RB, 256, 0, stream>>>(aj, mask, Sb, cnt);
        msum_kernel<<<RB * 32, 256, 0, stream>>>(Wl + 128 * 16, ai, bm[l + 1], Sb, cnt, mask,
                                                 (float*)X, 80, 64);
        gemm16_wmma<4, 80, 64, 80, 80, 4, 1><<<32, 256, 0, stream>>>(X, Wf[l + 1], bf[l + 1], Hout[l]);
    }

    // head hidden: z = relu(hB @ Wt1 + bt1) : (R x 64) @ (64 x 128)
    gemm16_wmma<4, 80, 128, 128, 64, 8, 1><<<64, 256, 0, stream>>>(hB, Wt1, bt1, zbuf);
    // final 7-wide projection + quaternion normalize
    head_final<<<(RTOT + 255) / 256, 256, 0, stream>>>(zbuf, Wt2, bt2, (float*)d_out);
    (void)in_sizes; (void)n_in; (void)out_size; (void)ws_size;
}